// CumulativeLayerNorm_85899346482
// MI455X (gfx1250) — compile-verified
//
#include <hip/hip_runtime.h>
#include <hip/hip_bf16.h>

// CDNA5 / gfx1250 cumulative layer norm.
// inputs: [B=8, K=8000, H=512] f32; gamma/beta: [1,512] f32; out same shape as inputs.

typedef __attribute__((ext_vector_type(2))) float v2f;
typedef __attribute__((ext_vector_type(8))) float v8f;

#ifndef CLN_H
#define CLN_H 512
#endif
#define CLN_EPS 1e-8f

// ---------------------------------------------------------------------------
// Kernel 1: per-frame sum and sum-of-squares over H, using the matrix unit.
// Each wave32 owns 16 consecutive frames (rows). For each K-chunk of 4 columns
// it builds the 16x4 f32 A tile (per ISA layout: lanes 0-15 hold K=0,1 in
// v0,v1; lanes 16-31 hold K=2,3) and does D += A x ones(4x16). Row sums land
// replicated across N, rows 0-7 in lane 0's 8 acc VGPRs, rows 8-15 in lane 16.
// ---------------------------------------------------------------------------
__global__ void cln_rowsums_wmma(const float* __restrict__ in,
                                 float* __restrict__ step_sum,
                                 float* __restrict__ step_ssq,
                                 int frames) {
    const int wave = threadIdx.x >> 5;          // 8 waves / block
    const int lane = threadIdx.x & 31;
    const long frame0 = ((long)blockIdx.x * 8 + wave) * 16;
    if (frame0 + 16 > frames) return;           // wave-uniform: EXEC stays all-1s

    const int row = lane & 15;                  // M index of this lane
    const int colbase = (lane >> 4) << 1;       // 0 for lanes 0-15, 2 for 16-31
    const float* rp = in + ((size_t)frame0 + row) * CLN_H + colbase;

    const v2f bones = {1.0f, 1.0f};             // B = ones => D[m,n] = rowsum(A[m])
    v8f acc_s = {0.f, 0.f, 0.f, 0.f, 0.f, 0.f, 0.f, 0.f};
    v8f acc_q = {0.f, 0.f, 0.f, 0.f, 0.f, 0.f, 0.f, 0.f};

#pragma unroll 4
    for (int kc = 0; kc < CLN_H / 4; ++kc) {
        float2 xv = *reinterpret_cast<const float2*>(rp + 4 * kc);
        v2f a  = {xv.x, xv.y};
        v2f a2 = {xv.x * xv.x, xv.y * xv.y};
        acc_s = __builtin_amdgcn_wmma_f32_16x16x4_f32(
            false, a,  false, bones, (short)0, acc_s, false, false);
        acc_q = __builtin_amdgcn_wmma_f32_16x16x4_f32(
            false, a2, false, bones, (short)0, acc_q, false, false);
    }

    // C/D layout: VGPR i, lanes 0-15 -> M=i ; lanes 16-31 -> M=i+8.
    if (lane == 0 || lane == 16) {
        const long mbase = frame0 + ((lane >> 4) << 3);
#pragma unroll
        for (int i = 0; i < 8; ++i) {
            step_sum[mbase + i] = acc_s[i];
            step_ssq[mbase + i] = acc_q[i];
        }
    }
}

// ---------------------------------------------------------------------------
// Kernel 2: per-batch inclusive scan of per-frame sums -> mean / inv_std.
// One wave32 per batch; Hillis-Steele scan with __shfl_up, serial carry.
// ---------------------------------------------------------------------------
__global__ void cln_scan(const float* __restrict__ step_sum,
                         const float* __restrict__ step_ssq,
                         float* __restrict__ mean_arr,
                         float* __restrict__ istd_arr,
                         int K) {
    const int b = blockIdx.x;
    const int lane = threadIdx.x;               // blockDim.x == 32
    const long base = (long)b * K;

    float carry_s = 0.f, carry_q = 0.f;
    for (int k0 = 0; k0 < K; k0 += 32) {
        const int k = k0 + lane;
        float s = (k < K) ? step_sum[base + k] : 0.f;
        float q = (k < K) ? step_ssq[base + k] : 0.f;
#pragma unroll
        for (int d = 1; d < 32; d <<= 1) {
            float s2 = __shfl_up(s, d, 32);
            float q2 = __shfl_up(q, d, 32);
            if (lane >= d) { s += s2; q += q2; }
        }
        const float cs = carry_s + s;
        const float cq = carry_q + q;
        if (k < K) {
            const float count = (float)(k + 1) * (float)CLN_H;
            const float m   = cs / count;
            const float var = cq / count - m * m;
            mean_arr[base + k] = m;
            istd_arr[base + k] = 1.0f / sqrtf(var + CLN_EPS);
        }
        carry_s += __shfl(s, 31, 32);
        carry_q += __shfl(q, 31, 32);
    }
}

// ---------------------------------------------------------------------------
// Kernel 3: out = gamma * (x - mean) * inv_std + beta, float4 streaming with
// CDNA5 global prefetch (global_prefetch_b8) one grid-stride ahead.
// ---------------------------------------------------------------------------
__global__ void cln_normalize(const float* __restrict__ in,
                              const float* __restrict__ gamma,
                              const float* __restrict__ beta,
                              const float* __restrict__ mean_arr,
                              const float* __restrict__ istd_arr,
                              float* __restrict__ out,
                              long total4) {
    const float4* __restrict__ in4 = reinterpret_cast<const float4*>(in);
    const float4* __restrict__ g4  = reinterpret_cast<const float4*>(gamma);
    const float4* __restrict__ b4  = reinterpret_cast<const float4*>(beta);
    float4* __restrict__ out4 = reinterpret_cast<float4*>(out);

    const long stride = (long)gridDim.x * blockDim.x;
    for (long i = (long)blockIdx.x * blockDim.x + threadIdx.x; i < total4; i += stride) {
        // Speculative prefetch of next iteration's line (silently dropped if OOB).
        __builtin_prefetch(in4 + i + stride, 0, 0);

        const long frame = i >> 7;              // H/4 == 128 float4 per frame
        const int  h4    = (int)(i & 127);

        const float4 x  = in4[i];
        const float  m  = mean_arr[frame];
        const float  is = istd_arr[frame];
        const float4 g  = g4[h4];
        const float4 bb = b4[h4];

        float4 r;
        r.x = g.x * ((x.x - m) * is) + bb.x;
        r.y = g.y * ((x.y - m) * is) + bb.y;
        r.z = g.z * ((x.z - m) * is) + bb.z;
        r.w = g.w * ((x.w - m) * is) + bb.w;
        out4[i] = r;
    }
}

// ---------------------------------------------------------------------------
extern "C" void kernel_launch(void* const* d_in, const int* in_sizes, int n_in,
                              void* d_out, int out_size, void* d_ws, size_t ws_size,
                              hipStream_t stream) {
    const float* in    = (const float*)d_in[0];
    const float* gamma = (const float*)d_in[1];
    const float* beta  = (const float*)d_in[2];
    float* out = (float*)d_out;

    const int H = CLN_H;                        // 512 per reference
    const int total = in_sizes[0];              // B*K*H
    const int frames = total / H;               // B*K = 64000
    const int B = 8;                            // per reference setup
    const int K = frames / B;                   // 8000

    float* step_sum = (float*)d_ws;
    float* step_ssq = step_sum + frames;
    float* mean_arr = step_ssq + frames;
    float* istd_arr = mean_arr + frames;        // 4*frames floats = 1 MB of ws

    // Pass 1: 16 frames per wave, 8 waves (256 threads) per block.
    const int frames_per_block = 16 * 8;
    const int g1 = (frames + frames_per_block - 1) / frames_per_block;
    cln_rowsums_wmma<<<g1, 256, 0, stream>>>(in, step_sum, step_ssq, frames);

    // Pass 2: one wave32 per batch.
    cln_scan<<<B, 32, 0, stream>>>(step_sum, step_ssq, mean_arr, istd_arr, K);

    // Pass 3: streaming normalize.
    const long total4 = (long)frames * (H / 4);
    cln_normalize<<<2048, 256, 0, stream>>>(in, gamma, beta, mean_arr, istd_arr,
                                            out, total4);
}